// DifferentiableBMCSimWrapper_22952305230379
// MI455X (gfx1250) — compile-verified
//
#include <hip/hip_runtime.h>
#include <math.h>

// ---------------------------------------------------------------------------
// Bloch-McConnell 2-pool CEST simulation on gfx1250 (MI455X).
//
// Strategy (see analysis): 7x7 augmented matrix-exponential propagator
// (removes the per-step linear solve), two instances per wave packed
// block-diagonally into one 16x16 WMMA tile, all 6x6/7x7 matmuls executed
// as chains of V_WMMA_F32_16X16X4_F32 with LDS-based operand re-layout.
// ---------------------------------------------------------------------------

typedef __attribute__((ext_vector_type(2))) float v2f;
typedef __attribute__((ext_vector_type(8))) float v8f;

#define TWO_PI_F   6.28318530717958647692f
#define GAMMA_UT   42.577f
#define R1A_C      (1.0f / 1.3f)
#define R2A_C      (1.0f / 0.075f)
#define R1B_C      1.0f
#define R2B_C      30.0f
#define KB_C       200.0f
#define FB_C       0.01f
#define KA_C       (FB_C * KB_C)
#define DWB_HZ_C   447.0f
#define DT_C       3.9e-05f
#define RF_FREQ_C  200.0f
#define N_RF_C     4
#define N_SAMP_C   128
#define N_OFF_C    32768
// Taylor(8) + 4 squarings, matrix pre-scaled by DT / 2^4.
#define HSC_C      (DT_C / 16.0f)

__global__ void bmc_init_out(unsigned int* __restrict__ o) {
    if (threadIdx.x < 2) o[threadIdx.x] = 0u;   // bits of +0.0f
}

__global__ void __launch_bounds__(32)
bmc_sim_kernel(const float* __restrict__ amps,      // [4*128]
               const float* __restrict__ phs,       // [4*128]
               const float* __restrict__ offsets,   // [32768]
               const float* __restrict__ m_init,    // [6]
               unsigned int* __restrict__ out_bits) // [2] float bits
{
    // One wave per block. Two offset-instances per wave, occupying
    // rows/cols [0,7) and [7,14) of the 16x16 tile (block diagonal).
    __shared__ float tA[256];   // augmented, scaled system matrix (col-major: [n*16+m])
    __shared__ float tT[256];   // staging tile for term / E re-layout
    __shared__ float xv[16];    // [mag;1] vectors: inst0 -> 0..6, inst1 -> 7..13

    const int lane = threadIdx.x;
    const int hi   = lane >> 4;      // wave half
    const int lid  = lane & 15;
    const int inst0 = blockIdx.x * 2;              // both instances share rf sign
    const int sign  = (inst0 >= N_OFF_C) ? 1 : 0;  // 0 -> +RF, 1 -> -RF
    const float rf  = sign ? -RF_FREQ_C : RF_FREQ_C;

    for (int i = lane; i < 256; i += 32) { tA[i] = 0.0f; tT[i] = 0.0f; }
    if (lane < 16) xv[lane] = 0.0f;
    __syncthreads();

    // ---- one-time constant entries of Ahat*h (two lanes, one instance each)
    if (lane < 2) {
        const int   inst = inst0 + lane;
        const float off  = offsets[inst & (N_OFF_C - 1)];
        const float dwa  = TWO_PI_F * (off - rf);
        const float dwb  = TWO_PI_F * (off + DWB_HZ_C - rf);
        const int   o    = 7 * lane;
        const float h    = HSC_C;
#define TW(m, n, v) tA[((n) + o) * 16 + ((m) + o)] = (v) * h
        TW(0,0, -(R2A_C + KA_C)); TW(1,1, -(R2A_C + KA_C)); TW(2,2, -(R1A_C + KA_C));
        TW(3,3, -(R2B_C + KB_C)); TW(4,4, -(R2B_C + KB_C)); TW(5,5, -(R1B_C + KB_C));
        TW(0,3, KB_C); TW(1,4, KB_C); TW(2,5, KB_C);
        TW(3,0, KA_C); TW(4,1, KA_C); TW(5,2, KA_C);
        TW(0,1,  dwa); TW(1,0, -dwa);
        TW(3,4,  dwb); TW(4,3, -dwb);
        TW(2,6, R1A_C); TW(5,6, R1B_C * FB_C);   // b-vector column (augmented)
#undef TW
        for (int i = 0; i < 6; ++i) xv[o + i] = m_init[i];
        xv[o + 6] = 1.0f;
    }

    // ---- per-step varying-entry assignment for lanes 0..15 (8 per instance)
    int em = 0, en = 0, epick = 0; float esgn = 0.0f;
    if (lane < 16) {
        const int e = lane & 7;
        // (m,n): (0,2)(1,2)(2,0)(2,1)(3,5)(4,5)(5,3)(5,4)
        em = e - (e > 2 ? 1 : 0) - (e > 6 ? 1 : 0);
        en = (e < 2) ? 2 : (e == 2) ? 0 : (e == 3) ? 1 : (e < 6) ? 5 : (e == 6) ? 3 : 4;
        esgn  = ((e + 1) & 2) ? 1.0f : -1.0f;     // -,+,+,-,-,+,+,-
        epick = e & 1;                            // 0 -> w1*sin, 1 -> w1*cos
        const int io = 7 * (lane >> 3);           // instance block offset
        em += io; en += io;
    }

    // phase accumulated per rf block (python-modulo semantics)
    float pd = fmodf(DT_C * N_SAMP_C * 360.0f * rf, 360.0f);
    if (pd < 0.0f) pd += 360.0f;
    const float pd_rad = pd * (3.14159265358979f / 180.0f);

    const int dbase = lid * 16 + 8 * hi;   // D-layout <-> LDS column-major base
    float acc = 0.0f;                      // lanes 0/1 hold the real maxima
    float acc_phase = 0.0f;

    for (int r = 0; r < N_RF_C; ++r) {
#pragma unroll 1
        for (int t = 0; t < N_SAMP_C; ++t) {
            const float amp = amps[r * N_SAMP_C + t];
            const float ph  = -phs[r * N_SAMP_C + t] - acc_phase;
            const float w1  = TWO_PI_F * GAMMA_UT * amp;
            float sp, cp;
            __sincosf(ph, &sp, &cp);
            const float ws = w1 * sp, wc = w1 * cp;

            __syncthreads();  // previous step's tA reads complete
            if (lane < 16) {
                const float v = epick ? wc : ws;
                tA[en * 16 + em] = esgn * v * HSC_C;
            }
            __syncthreads();

            // Ahat in A-layout (fixed operand of the whole Taylor chain)
            // and in B-layout (term_1).
            v2f aA[4], bT[4];
#pragma unroll
            for (int c = 0; c < 4; ++c) {
                const int k0 = 4 * c + 2 * hi;
                aA[c].x = tA[(k0 + 0) * 16 + lid];
                aA[c].y = tA[(k0 + 1) * 16 + lid];
                bT[c].x = tA[lid * 16 + k0];
                bT[c].y = tA[lid * 16 + k0 + 1];
            }

            // out = I + Ahat  (D-layout registers)
            v8f out;
#pragma unroll
            for (int i = 0; i < 8; ++i) {
                const int m = i + 8 * hi;
                out[i] = tA[dbase + i] + ((m == lid) ? 1.0f : 0.0f);
            }

            // Taylor k = 2..8 : term_k = (Ahat @ term_{k-1}) / k ; out += term_k
#pragma unroll
            for (int k = 2; k <= 8; ++k) {
                v8f d = {};
#pragma unroll
                for (int c = 0; c < 4; ++c)
                    d = __builtin_amdgcn_wmma_f32_16x16x4_f32(
                            false, aA[c], false, bT[c], (short)0, d, false, false);
                const float invk = 1.0f / (float)k;
#pragma unroll
                for (int i = 0; i < 8; ++i) { d[i] *= invk; out[i] += d[i]; }
                if (k < 8) {
                    __syncthreads();
#pragma unroll
                    for (int i = 0; i < 8; ++i) tT[dbase + i] = d[i];  // D -> LDS
                    __syncthreads();
#pragma unroll
                    for (int c = 0; c < 4; ++c) {                      // LDS -> B
                        const int k0 = 4 * c + 2 * hi;
                        bT[c].x = tT[lid * 16 + k0];
                        bT[c].y = tT[lid * 16 + k0 + 1];
                    }
                }
            }

            // 4 squarings: out = out @ out
#pragma unroll
            for (int s = 0; s < 4; ++s) {
                __syncthreads();
#pragma unroll
                for (int i = 0; i < 8; ++i) tT[dbase + i] = out[i];
                __syncthreads();
                v2f a2[4], b2[4];
#pragma unroll
                for (int c = 0; c < 4; ++c) {
                    const int k0 = 4 * c + 2 * hi;
                    a2[c].x = tT[(k0 + 0) * 16 + lid];
                    a2[c].y = tT[(k0 + 1) * 16 + lid];
                    b2[c].x = tT[lid * 16 + k0];
                    b2[c].y = tT[lid * 16 + k0 + 1];
                }
                v8f d = {};
#pragma unroll
                for (int c = 0; c < 4; ++c)
                    d = __builtin_amdgcn_wmma_f32_16x16x4_f32(
                            false, a2[c], false, b2[c], (short)0, d, false, false);
                out = d;
            }

            // propagate state: x = E @ x  (augmented row keeps x[6]=1)
            __syncthreads();
#pragma unroll
            for (int i = 0; i < 8; ++i) tT[dbase + i] = out[i];
            __syncthreads();
            float xn = 0.0f;
            if (lane < 14) {
                const int i  = lane;
                const int jb = (i < 7) ? 0 : 7;
#pragma unroll
                for (int j = 0; j < 7; ++j)
                    xn += tT[(jb + j) * 16 + i] * xv[jb + j];
            }
            __syncthreads();
            if (lane < 14) xv[lane] = xn;
            __syncthreads();
            if (lane < 2) {
                const int o = 7 * lane;
                const float tr = sqrtf(xv[o] * xv[o] + xv[o + 1] * xv[o + 1]);
                acc = fmaxf(acc, tr);
            }
        }
        acc_phase += pd_rad;
    }

    // non-negative float bits are order-preserving as uint
    if (lane < 2) atomicMax(&out_bits[sign], __float_as_uint(acc));
}

extern "C" void kernel_launch(void* const* d_in, const int* in_sizes, int n_in,
                              void* d_out, int out_size, void* d_ws, size_t ws_size,
                              hipStream_t stream) {
    (void)in_sizes; (void)n_in; (void)out_size; (void)d_ws; (void)ws_size;
    const float* amps    = (const float*)d_in[0];   // (4,128)
    const float* phs     = (const float*)d_in[1];   // (4,128)
    const float* offsets = (const float*)d_in[2];   // (32768,)
    const float* m_init  = (const float*)d_in[3];   // (6,)
    unsigned int* ob     = (unsigned int*)d_out;    // 2 floats (as bits)

    bmc_init_out<<<1, 32, 0, stream>>>(ob);
    // 65536 instances (2 signs x 32768 offsets), 2 per wave, 1 wave per block
    bmc_sim_kernel<<<dim3(N_OFF_C), dim3(32), 0, stream>>>(amps, phs, offsets, m_init, ob);
}